// MOLAttn_22144851378763
// MI455X (gfx1250) — compile-verified
//
#include <hip/hip_runtime.h>
#include <math.h>

// Problem constants (from reference)
#define BATCH 64
#define TDIM  2048
#define ENCD  512
#define ADIM  512
#define KMIX  5
#define G3    (3 * ADIM)    // 1536
#define XDIM  (ADIM + ENCD) // 1024

typedef __attribute__((ext_vector_type(2))) float v2f;
typedef __attribute__((ext_vector_type(4))) float v4f;
typedef __attribute__((ext_vector_type(8))) float v8f;

// ---------------------------------------------------------------------------
// x = concat(rnn_c, last_out) : (B, 1024)
// ---------------------------------------------------------------------------
__global__ void concat_x_kernel(const float* __restrict__ rnn_c,
                                const float* __restrict__ last_out,
                                float* __restrict__ x) {
    int idx = blockIdx.x * blockDim.x + threadIdx.x;   // B*XDIM = 65536
    int b = idx >> 10;
    int j = idx & 1023;
    x[idx] = (j < ADIM) ? rnn_c[b * ADIM + j] : last_out[b * ENCD + (j - ADIM)];
}

// ---------------------------------------------------------------------------
// WMMA GEMM helpers: 32-wide K chunk = 8 steps of {A, B0, B1} v2f fragments.
// Static register buffers only (no dynamic indexing -> no scratch).
// ---------------------------------------------------------------------------
__device__ __forceinline__ void ldchunk(v2f (&a)[8], v2f (&b0)[8], v2f (&b1)[8],
                                        const float* __restrict__ Ar,
                                        const float* __restrict__ W0,
                                        const float* __restrict__ W1, int k) {
#pragma unroll
    for (int u = 0; u < 8; ++u) {
        a[u]  = *(const v2f*)(Ar + k + 4 * u);
        b0[u] = *(const v2f*)(W0 + k + 4 * u);
        b1[u] = *(const v2f*)(W1 + k + 4 * u);
    }
}

__device__ __forceinline__ void mmchunk(v8f& acc0, v8f& acc1,
                                        const v2f (&a)[8], const v2f (&b0)[8],
                                        const v2f (&b1)[8]) {
#pragma unroll
    for (int u = 0; u < 8; ++u) {
        acc0 = __builtin_amdgcn_wmma_f32_16x16x4_f32(false, a[u], false, b0[u],
                                                     (short)0, acc0, false, false);
        acc1 = __builtin_amdgcn_wmma_f32_16x16x4_f32(false, a[u], false, b1[u],
                                                     (short)0, acc1, false, false);
    }
}

// ---------------------------------------------------------------------------
// fp32 WMMA GEMM: C(M,N) = A(M,Kd) @ W(N,Kd)^T + bias, optional tanh.
// One wave owns a 16x32 strip (1 M-tile x 2 N-tiles). K is walked in
// 32-wide chunks, double-buffered in registers: the next chunk's 24 b64
// loads are issued before the current chunk's 16 V_WMMA_F32_16X16X4_F32
// consume their (already-resident) operands -> partial s_wait_loadcnt,
// L2 latency hidden. Kd must be a multiple of 64 (even #chunks).
// Launch waves = (M/16)*(N/32); block = 128 threads (4 waves).
// ---------------------------------------------------------------------------
template <int ACT_TANH>
__global__ void wmma_gemm_nt2(const float* __restrict__ A,
                              const float* __restrict__ W,
                              const float* __restrict__ bias,
                              float* __restrict__ C,
                              int M, int N, int Kd) {
    const int wave = (blockIdx.x * blockDim.x + threadIdx.x) >> 5;
    const int lane = threadIdx.x & 31;
    const int strips = N >> 5;           // N/32
    const int tilesM = M >> 4;
    const int tm = wave / strips;
    const int sn = wave % strips;
    if (tm >= tilesM) return;            // wave-uniform -> EXEC stays all-ones

    const int half = lane >> 4;          // K sub-pair select (ISA 32-bit layout)
    const int ml   = lane & 15;          // row (A) / col (B) within tile

    const float* __restrict__ Ar = A + (size_t)(tm * 16 + ml) * Kd + half * 2;
    const float* __restrict__ W0 = W + (size_t)(sn * 32 +  0 + ml) * Kd + half * 2;
    const float* __restrict__ W1 = W + (size_t)(sn * 32 + 16 + ml) * Kd + half * 2;

    v2f aA[8], aB0[8], aB1[8];           // buffer A
    v2f bA[8], bB0[8], bB1[8];           // buffer B
    v8f acc0 = {}, acc1 = {};

    ldchunk(aA, aB0, aB1, Ar, W0, W1, 0);
    const int nch = Kd >> 5;             // 16 or 32 (even)
    for (int c = 0; c < nch; c += 2) {
        ldchunk(bA, bB0, bB1, Ar, W0, W1, (c + 1) << 5);
        mmchunk(acc0, acc1, aA, aB0, aB1);
        if (c + 2 < nch) ldchunk(aA, aB0, aB1, Ar, W0, W1, (c + 2) << 5);
        mmchunk(acc0, acc1, bA, bB0, bB1);
    }

    // D layout: VGPR v -> row (v + 8*half), col = tile_n*16 + ml
    const int n0 = sn * 32 + ml;
    const float bv0 = bias[n0];
    const float bv1 = bias[n0 + 16];
#pragma unroll
    for (int v = 0; v < 8; ++v) {
        const int m = tm * 16 + v + half * 8;
        float* __restrict__ Crow = C + (size_t)m * N + n0;
        float o0 = acc0[v] + bv0;
        float o1 = acc1[v] + bv1;
        if (ACT_TANH) { o0 = tanhf(o0); o1 = tanhf(o1); }
        Crow[0]  = o0;
        Crow[16] = o1;
    }
}

// ---------------------------------------------------------------------------
// GRU gate fusion: h_new = (1-z)*n + z*h   (PyTorch gate order r,z,n)
// ---------------------------------------------------------------------------
__global__ void gru_gates_kernel(const float* __restrict__ gi,
                                 const float* __restrict__ gh,
                                 const float* __restrict__ h,
                                 float* __restrict__ h_new) {
    int idx = blockIdx.x * blockDim.x + threadIdx.x;   // B*ADIM = 32768
    int b = idx >> 9;
    int j = idx & 511;
    const float* gib = gi + (size_t)b * G3;
    const float* ghb = gh + (size_t)b * G3;
    float ir = gib[j],            hr = ghb[j];
    float iz = gib[ADIM + j],     hz = ghb[ADIM + j];
    float in = gib[2 * ADIM + j], hn = ghb[2 * ADIM + j];
    float r = 1.0f / (1.0f + expf(-(ir + hr)));
    float z = 1.0f / (1.0f + expf(-(iz + hz)));
    float n = tanhf(in + r * hn);
    h_new[idx] = (1.0f - z) * n + z * h[idx];
}

// ---------------------------------------------------------------------------
// params(B,15) = hidden(B,512) @ W2(15,512)^T + b2   (N=15 -> scalar reduce)
// ---------------------------------------------------------------------------
__global__ void params_gemm_kernel(const float* __restrict__ hidden,
                                   const float* __restrict__ W2,
                                   const float* __restrict__ b2,
                                   float* __restrict__ params) {
    int b = blockIdx.x / (3 * KMIX);
    int j = blockIdx.x % (3 * KMIX);
    __shared__ float red[64];
    float s = 0.0f;
    const float* hb = hidden + (size_t)b * ADIM;
    const float* wj = W2 + (size_t)j * ADIM;
    for (int k = threadIdx.x; k < ADIM; k += 64) s += hb[k] * wj[k];
    red[threadIdx.x] = s;
    __syncthreads();
    for (int off = 32; off > 0; off >>= 1) {
        if (threadIdx.x < off) red[threadIdx.x] += red[threadIdx.x + off];
        __syncthreads();
    }
    if (threadIdx.x == 0) params[b * (3 * KMIX) + j] = red[0] + b2[j];
}

// ---------------------------------------------------------------------------
// Per-batch mixture params: mean=exp(p0)+prev, inv_s=exp(-p1), w=softmax(p2)
// ---------------------------------------------------------------------------
__global__ void mixture_kernel(const float* __restrict__ params,
                               const float* __restrict__ prev_means,
                               float* __restrict__ mix) {
    int b = blockIdx.x * blockDim.x + threadIdx.x;
    if (b >= BATCH) return;
    float mean[KMIX], invs[KMIX], lw[KMIX];
    float mx = -1e30f;
#pragma unroll
    for (int k = 0; k < KMIX; ++k) {
        float p0 = params[b * (3 * KMIX) + k * 3 + 0];
        float p1 = params[b * (3 * KMIX) + k * 3 + 1];
        float p2 = params[b * (3 * KMIX) + k * 3 + 2];
        mean[k] = expf(p0) + prev_means[b * KMIX + k];
        invs[k] = expf(-p1);           // 1/scale
        lw[k] = p2;
        mx = fmaxf(mx, p2);
    }
    float sum = 0.0f;
#pragma unroll
    for (int k = 0; k < KMIX; ++k) { lw[k] = expf(lw[k] - mx); sum += lw[k]; }
    float inv_sum = 1.0f / sum;
#pragma unroll
    for (int k = 0; k < KMIX; ++k) {
        mix[(b * KMIX + k) * 3 + 0] = mean[k];
        mix[(b * KMIX + k) * 3 + 1] = invs[k];
        mix[(b * KMIX + k) * 3 + 2] = lw[k] * inv_sum;
    }
}

// ---------------------------------------------------------------------------
// rnn_w[b,t] = sum_k w_k * (sigmoid((t+.5-m)*is) - sigmoid((t-.5-m)*is))
// ---------------------------------------------------------------------------
__global__ void rnnw_kernel(const float* __restrict__ mix,
                            float* __restrict__ rnn_w) {
    int idx = blockIdx.x * blockDim.x + threadIdx.x;   // B*T = 131072
    int b = idx >> 11;
    int t = idx & (TDIM - 1);
    float pos = (float)t;
    float acc = 0.0f;
#pragma unroll
    for (int k = 0; k < KMIX; ++k) {
        float mean = mix[(b * KMIX + k) * 3 + 0];
        float invs = mix[(b * KMIX + k) * 3 + 1];
        float w    = mix[(b * KMIX + k) * 3 + 2];
        float f1 = 1.0f / (1.0f + expf(-((pos + 0.5f - mean) * invs)));
        float f2 = 1.0f / (1.0f + expf(-((pos - 0.5f - mean) * invs)));
        acc += w * (f1 - f2);
    }
    rnn_w[idx] = acc;
}

// ---------------------------------------------------------------------------
// ctx[b,e] = sum_t rnn_w[b,t] * enc_z[b,t,e]  -- the roofline term (256 MB).
// Grid (B, ENC/64); 256 threads = 16 float4 e-lanes x 16 t-groups.
// rnn_w row staged in LDS; enc_z streamed with non-temporal b128 loads
// (read-once, 256MB > 192MB L2 -> keep it out of the cache's way).
// ---------------------------------------------------------------------------
__global__ void ctx_kernel(const float* __restrict__ rnn_w,
                           const float* __restrict__ enc_z,
                           float* __restrict__ ctx) {
    const int b  = blockIdx.x;
    const int e0 = blockIdx.y * 64;           // 64 floats = 16 float4 per block
    const int q  = threadIdx.x & 15;          // float4 lane within e-chunk
    const int tg = threadIdx.x >> 4;          // 0..15 t-group

    __shared__ float srw[TDIM];               // 8 KB: this batch's rnn_w row
    const float* __restrict__ rw = rnn_w + (size_t)b * TDIM;
    for (int t = threadIdx.x; t < TDIM; t += 256) srw[t] = rw[t];
    __syncthreads();

    const v4f* __restrict__ ez =
        (const v4f*)(enc_z + (size_t)b * TDIM * ENCD + e0) + q;

    v4f acc = {0.0f, 0.0f, 0.0f, 0.0f};
    for (int t = tg; t < TDIM; t += 16) {     // 128 b128 NT loads per thread
        v4f e = __builtin_nontemporal_load(&ez[(size_t)t * (ENCD / 4)]);
        float r = srw[t];
        acc.x += r * e.x;
        acc.y += r * e.y;
        acc.z += r * e.z;
        acc.w += r * e.w;
    }

    __shared__ v4f red[256];
    red[threadIdx.x] = acc;
    __syncthreads();
    if (tg == 0) {                            // 16 threads fold 16 partials each
        v4f s = red[q];
#pragma unroll
        for (int g = 1; g < 16; ++g) {
            v4f p = red[g * 16 + q];
            s.x += p.x; s.y += p.y; s.z += p.z; s.w += p.w;
        }
        *(v4f*)(ctx + (size_t)b * ENCD + e0 + q * 4) = s;
    }
}

// ---------------------------------------------------------------------------
// Host-side orchestration
// ---------------------------------------------------------------------------
extern "C" void kernel_launch(void* const* d_in, const int* in_sizes, int n_in,
                              void* d_out, int out_size, void* d_ws, size_t ws_size,
                              hipStream_t stream) {
    const float* enc_z      = (const float*)d_in[0];
    const float* rnn_c      = (const float*)d_in[1];
    const float* rnn_h      = (const float*)d_in[2];
    const float* last_out   = (const float*)d_in[3];
    const float* prev_means = (const float*)d_in[4];
    const float* W_ih       = (const float*)d_in[5];
    const float* W_hh       = (const float*)d_in[6];
    const float* b_ih       = (const float*)d_in[7];
    const float* b_hh       = (const float*)d_in[8];
    const float* W1         = (const float*)d_in[9];
    const float* b1         = (const float*)d_in[10];
    const float* W2         = (const float*)d_in[11];
    const float* b2         = (const float*)d_in[12];

    float* out = (float*)d_out;
    // Output layout (flat, return order): ctx | rnn_w | h_new
    float* ctx   = out;                               // B*ENC   = 32768
    float* rnn_w = out + BATCH * ENCD;                // B*T     = 131072
    float* h_new = out + BATCH * ENCD + BATCH * TDIM; // B*ADIM  = 32768

    // Workspace layout (floats), ~1.2 MB total
    float* ws = (float*)d_ws;
    float* x      = ws;                        // 64*1024  = 65536
    float* gi     = x      + BATCH * XDIM;     // 64*1536  = 98304
    float* gh     = gi     + BATCH * G3;       // 64*1536  = 98304
    float* hidden = gh     + BATCH * G3;       // 64*512   = 32768
    float* params = hidden + BATCH * ADIM;     // 64*15    = 960
    float* mix    = params + BATCH * 3 * KMIX; // 64*5*3   = 960

    // 1) x = concat(rnn_c, last_out)
    concat_x_kernel<<<(BATCH * XDIM) / 256, 256, 0, stream>>>(rnn_c, last_out, x);

    // 2) gi = x @ W_ih^T + b_ih : (64,1536), K=1024 -> 4*(1536/32)=192 waves
    wmma_gemm_nt2<0><<<48, 128, 0, stream>>>(x, W_ih, b_ih, gi, BATCH, G3, XDIM);

    // 3) gh = rnn_h @ W_hh^T + b_hh : (64,1536), K=512 -> 192 waves
    wmma_gemm_nt2<0><<<48, 128, 0, stream>>>(rnn_h, W_hh, b_hh, gh, BATCH, G3, ADIM);

    // 4) GRU gates -> h_new (written directly into d_out)
    gru_gates_kernel<<<(BATCH * ADIM) / 256, 256, 0, stream>>>(gi, gh, rnn_h, h_new);

    // 5) hidden = tanh(h_new @ W1^T + b1) : (64,512), K=512 -> 4*16=64 waves
    wmma_gemm_nt2<1><<<16, 128, 0, stream>>>(h_new, W1, b1, hidden, BATCH, ADIM, ADIM);

    // 6) params = hidden @ W2^T + b2 : N=15 (not WMMA-shaped, tiny)
    params_gemm_kernel<<<BATCH * 3 * KMIX, 64, 0, stream>>>(hidden, W2, b2, params);

    // 7) mixture parameters (means/inv_scales/softmax weights)
    mixture_kernel<<<1, 64, 0, stream>>>(params, prev_means, mix);

    // 8) rnn_w (logistic CDF differences) -> d_out
    rnnw_kernel<<<(BATCH * TDIM) / 256, 256, 0, stream>>>(mix, rnn_w);

    // 9) ctx = sum_t rnn_w * enc_z : 512 blocks streaming b128 NT loads
    ctx_kernel<<<dim3(BATCH, ENCD / 64), 256, 0, stream>>>(rnn_w, enc_z, ctx);

    (void)in_sizes; (void)n_in; (void)out_size; (void)ws_size;
}